// EdgeFeatures_32263794328104
// MI455X (gfx1250) — compile-verified
//
#include <hip/hip_runtime.h>

// ---------------------------------------------------------------------------
// EdgeFeatures for MI455X (gfx1250, wave32).
// Pipeline:  topk (dist + top-30)  ->  frames (O matrices)  ->  edge (features
// + 39->128 GEMM via V_WMMA_F32_16X16X4_F32 + layernorm + coalesced store).
// Output layout in d_out: [E : B*N*K*128 f32][E_idx as f32 : B*N*K] (indices
// <= 2047 are exactly representable in f32).
// Workspace: [E_idx i32 : NE][D_nb f32 : NE][O f32 : B*N*9]  (~4.5 MB).
// ---------------------------------------------------------------------------

#define KNB   30
#define NRBF  16
#define NIN   39
#define EF    128
#define NN    2048

typedef float v2f __attribute__((ext_vector_type(2)));
typedef float v8f __attribute__((ext_vector_type(8)));

__device__ __forceinline__ float sgnf(float x) {
    return (float)((x > 0.0f) - (x < 0.0f));   // jnp.sign semantics, sign(0)=0
}

// ------------------------- kernel 1: distances + top-K ----------------------
__global__ __launch_bounds__(256)
void topk_kernel(const float* __restrict__ X, const float* __restrict__ mask,
                 int* __restrict__ Eidx, float* __restrict__ Dnb,
                 float* __restrict__ EidxOut, int B, int N) {
    __shared__ float dist[NN];
    __shared__ float smax[256];
    __shared__ unsigned long long red[256];

    const int row = blockIdx.x;           // row = b*N + i
    const int b   = row / N;
    const int i   = row % N;
    const int tid = threadIdx.x;

    const float* Xb = X + (size_t)b * N * 3;
    const float xi = Xb[i * 3 + 0], yi = Xb[i * 3 + 1], zi = Xb[i * 3 + 2];
    const float mi = mask[(size_t)b * N + i];

    // pass 1: D = m2 * sqrt(|dX|^2 + 1e-6), track row max
    float lmax = 0.0f;
    for (int j = tid; j < N; j += 256) {
        float dx = Xb[j * 3 + 0] - xi;
        float dy = Xb[j * 3 + 1] - yi;
        float dz = Xb[j * 3 + 2] - zi;
        float m2 = mi * mask[(size_t)b * N + j];
        float D  = m2 * sqrtf(dx * dx + dy * dy + dz * dz + 1e-6f);
        dist[j] = D;
        lmax = fmaxf(lmax, D);
    }
    smax[tid] = lmax;
    __syncthreads();
    for (int off = 128; off > 0; off >>= 1) {
        if (tid < off) smax[tid] = fmaxf(smax[tid], smax[tid + off]);
        __syncthreads();
    }
    const float Dmax = smax[0];

    // pass 2: D_adj = D + (1 - m2) * Dmax
    for (int j = tid; j < N; j += 256) {
        float m2 = mi * mask[(size_t)b * N + j];
        dist[j] = dist[j] + (1.0f - m2) * Dmax;
    }
    __syncthreads();

    // registers hold this thread's 8 candidates
    float v[NN / 256];
#pragma unroll
    for (int s = 0; s < NN / 256; s++) v[s] = dist[tid + s * 256];

    // 30 iterative argmin extractions; key = (f32bits << 32) | idx
    // (D >= 0 so float bits are monotone; ties -> smallest index, as in top_k)
    for (int k = 0; k < KNB; k++) {
        unsigned long long best = 0xFFFFFFFFFFFFFFFFull;
#pragma unroll
        for (int s = 0; s < NN / 256; s++) {
            unsigned long long key =
                (((unsigned long long)__float_as_uint(v[s])) << 32) |
                (unsigned)(tid + s * 256);
            best = (key < best) ? key : best;
        }
        red[tid] = best;
        __syncthreads();
        for (int off = 128; off > 0; off >>= 1) {
            if (tid < off) {
                unsigned long long o = red[tid + off];
                if (o < red[tid]) red[tid] = o;
            }
            __syncthreads();
        }
        unsigned long long win = red[0];
        __syncthreads();                       // before red is reused
        int wj = (int)(win & 0xFFFFFFFFu);
        if (tid == 0) {
            size_t e = (size_t)row * KNB + k;
            Eidx[e]    = wj;
            Dnb[e]     = __uint_as_float((unsigned)(win >> 32));
            EidxOut[e] = (float)wj;
        }
        if ((wj & 255) == tid) v[wj >> 8] = __uint_as_float(0x7F800000u); // +inf
    }
}

// ------------------------- kernel 2: orientation frames ---------------------
__device__ __forceinline__ void norm3e(float& x, float& y, float& z) {
    float n = fmaxf(sqrtf(x * x + y * y + z * z), 1e-12f);
    x /= n; y /= n; z /= n;
}

__global__ __launch_bounds__(256)
void frames_kernel(const float* __restrict__ X, float* __restrict__ O,
                   int B, int N) {
    int idx = blockIdx.x * blockDim.x + threadIdx.x;
    if (idx >= B * N) return;
    int b = idx / N, n = idx % N;
    float* o = O + (size_t)idx * 9;
    if (n < 1 || n >= N - 2) {                  // jnp.pad ((1,2)) zeros
#pragma unroll
        for (int q = 0; q < 9; q++) o[q] = 0.0f;
        return;
    }
    const float* Xb = X + (size_t)b * N * 3;
    const int m = n - 1;
    float upx = Xb[(m + 1) * 3 + 0] - Xb[m * 3 + 0];
    float upy = Xb[(m + 1) * 3 + 1] - Xb[m * 3 + 1];
    float upz = Xb[(m + 1) * 3 + 2] - Xb[m * 3 + 2];
    norm3e(upx, upy, upz);
    float unx = Xb[(m + 2) * 3 + 0] - Xb[(m + 1) * 3 + 0];
    float uny = Xb[(m + 2) * 3 + 1] - Xb[(m + 1) * 3 + 1];
    float unz = Xb[(m + 2) * 3 + 2] - Xb[(m + 1) * 3 + 2];
    norm3e(unx, uny, unz);
    float nx = upy * unz - upz * uny;
    float ny = upz * unx - upx * unz;
    float nz = upx * uny - upy * unx;
    norm3e(nx, ny, nz);
    float bx = upx - unx, by = upy - uny, bz = upz - unz;
    norm3e(bx, by, bz);
    float cx = by * nz - bz * ny;
    float cy = bz * nx - bx * nz;
    float cz = bx * ny - by * nx;
    o[0] = bx; o[1] = by; o[2] = bz;
    o[3] = nx; o[4] = ny; o[5] = nz;
    o[6] = cx; o[7] = cy; o[8] = cz;
}

// ---------------- kernel 3: features + WMMA GEMM + layernorm ----------------
// One wave = one tile of 16 edges x 128 outputs. 128 threads = 4 tiles/block.
__global__ __launch_bounds__(128)
void edge_kernel(const float* __restrict__ X, const float* __restrict__ W,
                 const float* __restrict__ bias, const float* __restrict__ gamma,
                 const float* __restrict__ beta, const int* __restrict__ Eidx,
                 const float* __restrict__ Dnb, const float* __restrict__ O,
                 float* __restrict__ out, int B, int N) {
    __shared__ float Wl[EF * 41];        // W[n][k], stride 41 (bank-clean)
    __shared__ float bgl[3 * EF];        // bias | gamma | beta
    __shared__ float wbuf[4][16 * 132];  // per-wave: 16x41 feats, then 16x132 transpose

    const int tid = threadIdx.x;

    // ---- stage W / bias / gamma / beta ----
    for (int t = tid; t < EF * NIN; t += 128)
        Wl[(t / NIN) * 41 + (t % NIN)] = W[t];
    if (tid < EF) {
        Wl[tid * 41 + 39]  = 0.0f;       // K pad 39->40
        bgl[tid]           = bias[tid];
        bgl[EF + tid]      = gamma[tid];
        bgl[2 * EF + tid]  = beta[tid];
    }
    __syncthreads();

    const int wave = tid >> 5, lane = tid & 31;
    float* feat = wbuf[wave];
    const int tileBase = (blockIdx.x * 4 + wave) * 16;

    const int    e    = lane & 15;
    const size_t ed   = (size_t)tileBase + e;
    const int    bIdx = (int)(ed / ((size_t)N * KNB));
    const int    nI   = (int)((ed / KNB) % (size_t)N);
    const int    j    = Eidx[ed];

    if (lane < 16) {
        // ---- positional encodings (16) + RBF (16) ----
        float Dv = Dnb[ed];
        float d  = (float)(j - nI);
        const float freqs[8] = {1.0f, 0.31622776601683794f, 0.1f,
                                0.031622776601683794f, 0.01f,
                                0.0031622776601683794f, 0.001f,
                                0.00031622776601683794f};   // 10000^(-p/8)
#pragma unroll
        for (int p = 0; p < 8; p++) {
            float ang = d * freqs[p];
            feat[e * 41 + p]     = cosf(ang);
            feat[e * 41 + 8 + p] = sinf(ang);
        }
#pragma unroll
        for (int f = 0; f < NRBF; f++) {
            float mu = (20.0f / 15.0f) * (float)f;   // linspace(0,20,16)
            float t  = (Dv - mu) * 0.8f;             // sigma = 1.25
            feat[e * 41 + 16 + f] = expf(-t * t);
        }
    } else {
        // ---- orientation features: dU (3) + quaternion (4) ----
        const float* Oi = O + ((size_t)bIdx * N + nI) * 9;
        const float* Oj = O + ((size_t)bIdx * N + j) * 9;
        const float* Xi = X + ((size_t)bIdx * N + nI) * 3;
        const float* Xj = X + ((size_t)bIdx * N + j) * 3;
        float oi[9], oj[9];
#pragma unroll
        for (int q = 0; q < 9; q++) { oi[q] = Oi[q]; oj[q] = Oj[q]; }
        float dx = Xj[0] - Xi[0], dy = Xj[1] - Xi[1], dz = Xj[2] - Xi[2];
        float u0 = oi[0] * dx + oi[1] * dy + oi[2] * dz;
        float u1 = oi[3] * dx + oi[4] * dy + oi[5] * dz;
        float u2 = oi[6] * dx + oi[7] * dy + oi[8] * dz;
        norm3e(u0, u1, u2);
        float R[3][3];
#pragma unroll
        for (int a = 0; a < 3; a++)
#pragma unroll
            for (int l = 0; l < 3; l++)          // R = O^T * O_nb
                R[a][l] = oi[0 + a] * oj[0 + l] + oi[3 + a] * oj[3 + l] +
                          oi[6 + a] * oj[6 + l];
        float Rxx = R[0][0], Ryy = R[1][1], Rzz = R[2][2];
        float m0 = 0.5f * sqrtf(fabsf(1.0f + Rxx - Ryy - Rzz));
        float m1 = 0.5f * sqrtf(fabsf(1.0f - Rxx + Ryy - Rzz));
        float m2 = 0.5f * sqrtf(fabsf(1.0f - Rxx - Ryy + Rzz));
        float qx = sgnf(R[2][1] - R[1][2]) * m0;
        float qy = sgnf(R[0][2] - R[2][0]) * m1;
        float qz = sgnf(R[1][0] - R[0][1]) * m2;
        float qw = 0.5f * sqrtf(fmaxf(1.0f + Rxx + Ryy + Rzz, 0.0f));
        float qn = fmaxf(sqrtf(qx * qx + qy * qy + qz * qz + qw * qw), 1e-12f);
        feat[e * 41 + 32] = u0;  feat[e * 41 + 33] = u1;  feat[e * 41 + 34] = u2;
        feat[e * 41 + 35] = qx / qn;  feat[e * 41 + 36] = qy / qn;
        feat[e * 41 + 37] = qz / qn;  feat[e * 41 + 38] = qw / qn;
        feat[e * 41 + 39] = 0.0f;    // K pad
    }
    __syncthreads();

    // ---- GEMM: 16x128 = (16x40) @ (40x128) via 8 tiles x 10 WMMA f32 k=4 ----
    // A 16x4 f32 layout (ISA 7.12.2): lanes 0-15 K={4s,4s+1}, lanes 16-31 K={4s+2,4s+3}
    const int mrow = lane & 15;
    const int koff = (lane >> 4) << 1;            // 0 or 2
    v2f a[10];
#pragma unroll
    for (int s = 0; s < 10; s++) {
        a[s].x = feat[mrow * 41 + s * 4 + koff];
        a[s].y = feat[mrow * 41 + s * 4 + koff + 1];
    }
    v8f c[8];
#pragma unroll
    for (int t = 0; t < 8; t++) {
        v8f acc = {0.f, 0.f, 0.f, 0.f, 0.f, 0.f, 0.f, 0.f};
        const float* wrow = &Wl[(t * 16 + mrow) * 41];   // B[k][n] = W[n][k]
#pragma unroll
        for (int s = 0; s < 10; s++) {
            v2f bb;
            bb.x = wrow[s * 4 + koff];
            bb.y = wrow[s * 4 + koff + 1];
            acc = __builtin_amdgcn_wmma_f32_16x16x4_f32(
                false, a[s], false, bb, (short)0, acc, false, false);
        }
        c[t] = acc;
    }

    // ---- bias + layernorm (two-pass, half-wave shfl reductions) ----
#pragma unroll
    for (int t = 0; t < 8; t++) {
        float bb = bgl[t * 16 + mrow];
#pragma unroll
        for (int v = 0; v < 8; v++) c[t][v] += bb;
    }
    float mu[8], inv[8];
#pragma unroll
    for (int v = 0; v < 8; v++) {
        float s = 0.0f;
#pragma unroll
        for (int t = 0; t < 8; t++) s += c[t][v];
        s += __shfl_xor(s, 1, 32); s += __shfl_xor(s, 2, 32);
        s += __shfl_xor(s, 4, 32); s += __shfl_xor(s, 8, 32);
        mu[v] = s * (1.0f / 128.0f);
    }
#pragma unroll
    for (int v = 0; v < 8; v++) {
        float s = 0.0f;
#pragma unroll
        for (int t = 0; t < 8; t++) {
            float dv = c[t][v] - mu[v];
            s += dv * dv;
        }
        s += __shfl_xor(s, 1, 32); s += __shfl_xor(s, 2, 32);
        s += __shfl_xor(s, 4, 32); s += __shfl_xor(s, 8, 32);
        inv[v] = rsqrtf(s * (1.0f / 128.0f) + 1e-5f);
    }

    // ---- transpose through LDS (stride 132, bank-clean) + float4 stores ----
    __syncthreads();
    const int rowOff = (lane >> 4) << 3;          // C rows: lanes>=16 hold M+8
#pragma unroll
    for (int t = 0; t < 8; t++) {
        float g  = bgl[EF + t * 16 + mrow];
        float be = bgl[2 * EF + t * 16 + mrow];
#pragma unroll
        for (int v = 0; v < 8; v++) {
            float y = (c[t][v] - mu[v]) * inv[v] * g + be;
            feat[(v + rowOff) * 132 + t * 16 + mrow] = y;
        }
    }
    __syncthreads();
#pragma unroll
    for (int m = 0; m < 16; m++) {
        const float* src = &feat[m * 132 + lane * 4];
        float4 val = make_float4(src[0], src[1], src[2], src[3]);
        *(float4*)(out + ((size_t)(tileBase + m)) * EF + lane * 4) = val;
    }
}

// ---------------------------------------------------------------------------
extern "C" void kernel_launch(void* const* d_in, const int* in_sizes, int n_in,
                              void* d_out, int out_size, void* d_ws, size_t ws_size,
                              hipStream_t stream) {
    const float* X     = (const float*)d_in[0];
    const float* mask  = (const float*)d_in[1];
    const float* W     = (const float*)d_in[2];
    const float* bias  = (const float*)d_in[3];
    const float* gamma = (const float*)d_in[4];
    const float* beta  = (const float*)d_in[5];

    const int N = NN;
    const int B = in_sizes[1] / N;                // mask is B*N
    const size_t NE = (size_t)B * N * KNB;

    int*   Eidx = (int*)d_ws;
    float* Dnb  = (float*)((char*)d_ws + NE * sizeof(int));
    float* O    = (float*)((char*)d_ws + NE * (sizeof(int) + sizeof(float)));

    float* out     = (float*)d_out;
    float* EidxOut = out + NE * EF;               // E_idx (as f32) after E

    topk_kernel<<<B * N, 256, 0, stream>>>(X, mask, Eidx, Dnb, EidxOut, B, N);
    frames_kernel<<<(B * N + 255) / 256, 256, 0, stream>>>(X, O, B, N);

    const int tiles = (int)(NE / 16);             // 30720, exact
    edge_kernel<<<tiles / 4, 128, 0, stream>>>(X, W, bias, gamma, beta,
                                               Eidx, Dnb, O, out, B, N);
}